// DGCNNModel_21775484191346
// MI455X (gfx1250) — compile-verified
//
#include <hip/hip_runtime.h>
#include <math.h>

// ---------------- problem constants (match reference) ----------------
#define F_IN   128
#define B_GR   500
#define NPG    200
#define K_TOP  30
#define NPOS   (B_GR * K_TOP)   // 15000 conv5 rows
#define KPAD   100              // 97 padded to multiple of 4

typedef __attribute__((ext_vector_type(2))) float v2f;
typedef __attribute__((ext_vector_type(8))) float v8f;

// ---------------- utility kernels ----------------
__global__ void k_zero(float* p, int n) {
  int i = blockIdx.x * blockDim.x + threadIdx.x;
  if (i < n) p[i] = 0.f;
}

__global__ void k_degree(const int* __restrict__ dst, float* __restrict__ deg, int e) {
  int i = blockIdx.x * blockDim.x + threadIdx.x;
  if (i < e) atomicAdd(&deg[dst[i]], 1.0f);
}

__global__ void k_dis(float* d, int n) {  // deg -> 1/sqrt(deg+1) in place
  int i = blockIdx.x * blockDim.x + threadIdx.x;
  if (i < n) d[i] = rsqrtf(d[i] + 1.0f);
}

// ---------------- generic f32 WMMA GEMM: out[M,ncols] = A[M,K] @ W[K,ncols] ----------------
// K multiple of 4, ncols multiple of 32. Block = 256 (8 waves), wave tile = 16 rows x 32 cols.
// LDS holds W[:, col0:col0+32] in k-pair-interleaved form so each B fragment is one b64 load:
//   sW[((k>>1)*32 + n)*2 + (k&1)] = W[k][col0+n]
__global__ __launch_bounds__(256)
void k_gemm_wmma(const float* __restrict__ A, const float* __restrict__ W,
                 float* __restrict__ out, int M, int K, int ncols) {
  __shared__ float sW[352 * 32];            // max K = 352 -> 44 KB LDS
  const int tid  = threadIdx.x;
  const int lane = tid & 31;
  const int wave = tid >> 5;
  const int col0 = blockIdx.y * 32;

  for (int idx = tid; idx < K * 32; idx += 256) {
    int k = idx >> 5, c = idx & 31;
    sW[(((k >> 1) * 32 + c) << 1) | (k & 1)] = W[(size_t)k * ncols + col0 + c];
  }
  __syncthreads();

  const int row0 = blockIdx.x * 128 + wave * 16;
  const int nlo  = lane & 15;
  const int mrow = row0 + nlo;
  const int ksel = (lane < 16) ? 0 : 2;     // ISA A-layout: lanes 16-31 carry K+2,K+3
  // A row m only feeds D row m; OOB rows are dropped by the store guard,
  // so clamp the pointer (stay in-bounds) and load unconditionally.
  const int mclamp = (mrow < M) ? mrow : (M - 1);
  const float* arow = A + (size_t)mclamp * K + ksel;
  const float* brow0 = sW + ((size_t)(ksel >> 1) * 32 + nlo) * 2;
  const float* brow1 = brow0 + 32;          // +16 columns -> +32 floats

  v8f acc0 = {};
  v8f acc1 = {};
  #pragma unroll 4
  for (int kk = 0; kk < K; kk += 4) {
    v2f a  = *(const v2f*)(arow + kk);                    // one global_load_b64
    v2f b0 = *(const v2f*)(brow0 + (size_t)(kk >> 1) * 64); // one ds_load_b64
    v2f b1 = *(const v2f*)(brow1 + (size_t)(kk >> 1) * 64);
    acc0 = __builtin_amdgcn_wmma_f32_16x16x4_f32(false, a, false, b0, (short)0, acc0, false, false);
    acc1 = __builtin_amdgcn_wmma_f32_16x16x4_f32(false, a, false, b1, (short)0, acc1, false, false);
  }

  const int mbase = row0 + ((lane >= 16) ? 8 : 0);   // C/D layout: VGPR v -> rows v / v+8
  for (int v = 0; v < 8; ++v) {
    int r = mbase + v;
    if (r < M) {
      out[(size_t)r * ncols + col0 + nlo]      = acc0[v];
      out[(size_t)r * ncols + col0 + 16 + nlo] = acc1[v];
    }
  }
}

// ---------------- GCN edge aggregation ----------------
// 32-feature layers: lane = feature, coalesced 128B gather + scattered f32 atomics (L2-resident).
__global__ void k_edge_agg32(const int* __restrict__ src, const int* __restrict__ dst,
                             const float* __restrict__ dis, const float* __restrict__ h,
                             float* __restrict__ agg, int e) {
  long long t = (long long)blockIdx.x * blockDim.x + threadIdx.x;
  int ei = (int)(t >> 5);
  int f  = (int)(t & 31);
  if (ei < e) {
    int s = src[ei], d = dst[ei];
    float norm = dis[s] * dis[d];
    atomicAdd(&agg[(size_t)d * 32 + f], h[(size_t)s * 32 + f] * norm);
  }
}

__global__ void k_edge_agg1(const int* __restrict__ src, const int* __restrict__ dst,
                            const float* __restrict__ dis, const float* __restrict__ h,
                            float* __restrict__ agg, int e) {
  int ei = blockIdx.x * blockDim.x + threadIdx.x;
  if (ei < e) {
    int s = src[ei], d = dst[ei];
    atomicAdd(&agg[d], h[s] * dis[s] * dis[d]);
  }
}

__global__ void k_finish32(const float* __restrict__ agg, const float* __restrict__ h,
                           const float* __restrict__ dis, const float* __restrict__ bias,
                           float* __restrict__ xo, int n) {
  long long i = (long long)blockIdx.x * blockDim.x + threadIdx.x;
  if (i < (long long)n * 32) {
    int row = (int)(i >> 5), f = (int)(i & 31);
    float d2 = dis[row] * dis[row];
    xo[i] = tanhf(agg[i] + h[i] * d2 + bias[f]);
  }
}

__global__ void k_h4(const float* __restrict__ x3, const float* __restrict__ W4,
                     float* __restrict__ h4, int n) {
  int i = blockIdx.x * blockDim.x + threadIdx.x;
  if (i < n) {
    float s = 0.f;
    const float* r = x3 + (size_t)i * 32;
    #pragma unroll
    for (int k = 0; k < 32; ++k) s += r[k] * W4[k];
    h4[i] = s;
  }
}

__global__ void k_finish1(const float* __restrict__ agg, const float* __restrict__ h,
                          const float* __restrict__ dis, const float* __restrict__ b4,
                          float* __restrict__ xo, int n) {
  int i = blockIdx.x * blockDim.x + threadIdx.x;
  if (i < n) xo[i] = tanhf(agg[i] + h[i] * dis[i] * dis[i] + b4[0]);
}

// ---------------- global sort pool (top-30 by x4, descending, stable) ----------------
__global__ void k_sortpool(const float* __restrict__ x4, int* __restrict__ topk) {
  __shared__ float v[NPG];
  int b = blockIdx.x, tid = threadIdx.x;
  for (int i = tid; i < NPG; i += blockDim.x) v[i] = x4[b * NPG + i];
  __syncthreads();
  if (tid < NPG) {
    float mv = v[tid];
    int rank = 0;
    for (int j = 0; j < NPG; ++j) {
      float vj = v[j];
      rank += (vj > mv) || (vj == mv && j < tid);   // stable descending rank
    }
    if (rank < K_TOP) topk[b * K_TOP + rank] = b * NPG + tid;
  }
}

// gather concatenated features [x1|x2|x3|x4] into padded [NPOS, 100]
__global__ void k_gather(const int* __restrict__ topk,
                         const float* __restrict__ x1, const float* __restrict__ x2,
                         const float* __restrict__ x3, const float* __restrict__ x4,
                         float* __restrict__ tf) {
  int idx = blockIdx.x * blockDim.x + threadIdx.x;
  if (idx < NPOS * KPAD) {
    int r = idx / KPAD, i = idx % KPAD;
    int node = topk[r];
    float val;
    if (i < 32)       val = x1[(size_t)node * 32 + i];
    else if (i < 64)  val = x2[(size_t)node * 32 + i - 32];
    else if (i < 96)  val = x3[(size_t)node * 32 + i - 64];
    else if (i == 96) val = x4[node];
    else              val = 0.f;
    tf[idx] = val;
  }
}

// ---------------- weight repacks for the WMMA GEMMs ----------------
__global__ void k_w5p(const float* __restrict__ w5, float* __restrict__ o) {  // [16,97] -> [100,32] padded
  int idx = blockIdx.x * blockDim.x + threadIdx.x;
  if (idx < KPAD * 32) {
    int k = idx / 32, oc = idx % 32;
    o[idx] = (oc < 16 && k < 97) ? w5[oc * 97 + k] : 0.f;
  }
}
__global__ void k_w6p(const float* __restrict__ w6, float* __restrict__ o) {  // [32,80] -> [80,32]
  int idx = blockIdx.x * blockDim.x + threadIdx.x;
  if (idx < 80 * 32) {
    int k = idx / 32, oc = idx % 32;
    o[idx] = w6[oc * 80 + k];
  }
}
__global__ void k_wfc1(const float* __restrict__ w, float* __restrict__ o) {  // [128,352] -> [352,128]
  int idx = blockIdx.x * blockDim.x + threadIdx.x;
  if (idx < 352 * 128) {
    int k = idx / 128, c = idx % 128;
    o[idx] = w[c * 352 + k];
  }
}

// ---------------- conv head finishers ----------------
// conv5 bias+relu fused with MaxPool1d(2,2): o5 raw is [NPOS, 32] (cols 0..15 valid)
__global__ void k_pool(const float* __restrict__ o5, const float* __restrict__ b5,
                       float* __restrict__ pool) {
  int idx = blockIdx.x * blockDim.x + threadIdx.x;
  if (idx < B_GR * 16 * 15) {
    int b = idx / 240, rem = idx % 240;
    int ic = rem / 15, j = rem % 15;
    int r0 = b * K_TOP + 2 * j;
    float a = fmaxf(o5[(size_t)r0 * 32 + ic] + b5[ic], 0.f);
    float c = fmaxf(o5[(size_t)(r0 + 1) * 32 + ic] + b5[ic], 0.f);
    pool[b * 240 + ic * 15 + j] = fmaxf(a, c);
  }
}

__global__ void k_im2col(const float* __restrict__ pool, float* __restrict__ im2) {
  int idx = blockIdx.x * blockDim.x + threadIdx.x;
  if (idx < B_GR * 11 * 80) {
    int r = idx / 80, kk = idx % 80;
    int b = r / 11, j = r % 11;
    int ic = kk / 5, t = kk % 5;
    im2[idx] = pool[b * 240 + ic * 15 + j + t];
  }
}

// conv6 bias+relu + flatten into [B, 352] with layout b*352 + oc*11 + j
__global__ void k_c6fin(const float* __restrict__ o6, const float* __restrict__ b6,
                        float* __restrict__ hf) {
  int idx = blockIdx.x * blockDim.x + threadIdx.x;
  if (idx < B_GR * 352) {
    int b = idx / 352, rem = idx % 352;
    int oc = rem / 11, j = rem % 11;
    hf[idx] = fmaxf(o6[(size_t)(b * 11 + j) * 32 + oc] + b6[oc], 0.f);
  }
}

__global__ void k_fc1relu(float* __restrict__ f, const float* __restrict__ b) {
  int i = blockIdx.x * blockDim.x + threadIdx.x;
  if (i < B_GR * 128) f[i] = fmaxf(f[i] + b[i & 127], 0.f);
}

__global__ void k_fc2(const float* __restrict__ f, const float* __restrict__ w,
                      const float* __restrict__ b, float* __restrict__ out) {
  int idx = blockIdx.x * blockDim.x + threadIdx.x;
  if (idx < B_GR * 10) {
    int bb = idx / 10, c = idx % 10;
    float s = b[c];
    const float* fr = f + bb * 128;
    const float* wr = w + c * 128;
    #pragma unroll 4
    for (int k = 0; k < 128; ++k) s += fr[k] * wr[k];
    out[idx] = s;
  }
}

// ---------------- host-side orchestration ----------------
#define L1(kern, n, ...) kern<<<dim3((unsigned)(((long long)(n) + 255) / 256)), dim3(256), 0, stream>>>(__VA_ARGS__)

extern "C" void kernel_launch(void* const* d_in, const int* in_sizes, int n_in,
                              void* d_out, int out_size, void* d_ws, size_t ws_size,
                              hipStream_t stream) {
  const float* x   = (const float*)d_in[0];
  const int*   ei  = (const int*)d_in[1];
  const int N = in_sizes[0] / F_IN;
  const int E = in_sizes[1] / 2;
  const int* src = ei;
  const int* dst = ei + E;
  const float* W1 = (const float*)d_in[3];  const float* b1 = (const float*)d_in[4];
  const float* W2 = (const float*)d_in[5];  const float* b2 = (const float*)d_in[6];
  const float* W3 = (const float*)d_in[7];  const float* b3 = (const float*)d_in[8];
  const float* W4 = (const float*)d_in[9];  const float* b4 = (const float*)d_in[10];
  const float* c5w = (const float*)d_in[11]; const float* c5b = (const float*)d_in[12];
  const float* c6w = (const float*)d_in[13]; const float* c6b = (const float*)d_in[14];
  const float* f1w = (const float*)d_in[15]; const float* f1b = (const float*)d_in[16];
  const float* f2w = (const float*)d_in[17]; const float* f2b = (const float*)d_in[18];

  float* ws = (float*)d_ws;
  size_t off = 0;
  auto take = [&](size_t n) { float* p = ws + off; off += (n + 1023) & ~(size_t)1023; return p; };
  float* p_dis  = take(N);
  float* p_h    = take((size_t)N * 32);
  float* p_agg  = take((size_t)N * 32);     // layer 4 reuses first N floats
  float* p_x1   = take((size_t)N * 32);
  float* p_x2   = take((size_t)N * 32);
  float* p_x3   = take((size_t)N * 32);
  float* p_x4   = take(N);
  float* p_h4   = take(N);
  int*   p_topk = (int*)take(B_GR * K_TOP);
  float* p_tf   = take((size_t)NPOS * KPAD);
  float* p_w5   = take(KPAD * 32);
  float* p_o5   = take((size_t)NPOS * 32);
  float* p_pool = take(B_GR * 16 * 15);
  float* p_im2  = take((size_t)B_GR * 11 * 80);
  float* p_w6   = take(80 * 32);
  float* p_o6   = take((size_t)B_GR * 11 * 32);
  float* p_hf   = take(B_GR * 352);
  float* p_wfc1 = take(352 * 128);
  float* p_fc1  = take(B_GR * 128);

  auto gemm = [&](const float* A, const float* W, float* o, int M, int K, int ncols) {
    dim3 g((M + 127) / 128, ncols / 32);
    k_gemm_wmma<<<g, dim3(256), 0, stream>>>(A, W, o, M, K, ncols);
  };

  // degree -> dis = rsqrt(deg + 1)
  L1(k_zero, N, p_dis, N);
  L1(k_degree, E, dst, p_dis, E);
  L1(k_dis, N, p_dis, N);

  // weight repacks (independent of graph work)
  L1(k_w5p, KPAD * 32, c5w, p_w5);
  L1(k_w6p, 80 * 32, c6w, p_w6);
  L1(k_wfc1, 352 * 128, f1w, p_wfc1);

  // GCN layer 1: [N,128]@[128,32]
  gemm(x, W1, p_h, N, 128, 32);
  L1(k_zero, N * 32, p_agg, N * 32);
  L1(k_edge_agg32, (long long)E * 32, src, dst, p_dis, p_h, p_agg, E);
  L1(k_finish32, (long long)N * 32, p_agg, p_h, p_dis, b1, p_x1, N);

  // GCN layer 2
  gemm(p_x1, W2, p_h, N, 32, 32);
  L1(k_zero, N * 32, p_agg, N * 32);
  L1(k_edge_agg32, (long long)E * 32, src, dst, p_dis, p_h, p_agg, E);
  L1(k_finish32, (long long)N * 32, p_agg, p_h, p_dis, b2, p_x2, N);

  // GCN layer 3
  gemm(p_x2, W3, p_h, N, 32, 32);
  L1(k_zero, N * 32, p_agg, N * 32);
  L1(k_edge_agg32, (long long)E * 32, src, dst, p_dis, p_h, p_agg, E);
  L1(k_finish32, (long long)N * 32, p_agg, p_h, p_dis, b3, p_x3, N);

  // GCN layer 4 (width 1)
  L1(k_h4, N, p_x3, W4, p_h4, N);
  L1(k_zero, N, p_agg, N);
  L1(k_edge_agg1, E, src, dst, p_dis, p_h4, p_agg, E);
  L1(k_finish1, N, p_agg, p_h4, p_dis, b4, p_x4, N);

  // sort pool + gather
  k_sortpool<<<dim3(B_GR), dim3(256), 0, stream>>>(p_x4, p_topk);
  L1(k_gather, NPOS * KPAD, p_topk, p_x1, p_x2, p_x3, p_x4, p_tf);

  // conv5 as GEMM [15000,100]@[100,32], bias+relu+maxpool fused
  gemm(p_tf, p_w5, p_o5, NPOS, KPAD, 32);
  L1(k_pool, B_GR * 16 * 15, p_o5, c5b, p_pool);

  // conv6 as im2col GEMM [5500,80]@[80,32]
  L1(k_im2col, B_GR * 11 * 80, p_pool, p_im2);
  gemm(p_im2, p_w6, p_o6, B_GR * 11, 80, 32);
  L1(k_c6fin, B_GR * 352, p_o6, c6b, p_hf);

  // fc1 [500,352]@[352,128] + relu
  gemm(p_hf, p_wfc1, p_fc1, B_GR, 352, 128);
  L1(k_fc1relu, B_GR * 128, p_fc1, f1b);

  // fc2 -> [500,10]
  L1(k_fc2, B_GR * 10, p_fc1, f2w, f2b, (float*)d_out);
}